// LowRankMaskedSynapse_20349555048715
// MI455X (gfx1250) — compile-verified
//
#include <hip/hip_runtime.h>
#include <hip/hip_bf16.h>

// y = (spikes @ V) @ U^T
// spikes:[512,32768] f32, V:[32768,64] f32, U:[32768,64] f32 -> y:[512,32768] f32
// CSR mask inputs are unused by the reference.

#define N_PRE   32768
#define N_POST  32768
#define RANK    64
#define BATCH   512
#define KCHUNK  1024          // K elements per stage-1 block
#define LDS_STRIDE 1028       // 1024 + 4 dword pad -> conflict-free A-fragment reads

typedef __attribute__((ext_vector_type(2))) float v2f;
typedef __attribute__((ext_vector_type(8))) float v8f;

__device__ __forceinline__ v8f wmma_f32_16x16x4(v2f a, v2f b, v8f c) {
  // 8 args: (neg_a, A, neg_b, B, c_mod, C, reuse_a, reuse_b)
  return __builtin_amdgcn_wmma_f32_16x16x4_f32(
      false, a, false, b, (short)0, c, false, false);
}

// ---------------------------------------------------------------------------
// Kernel 0: zero the z accumulator (512 x 64 f32) in workspace.
// ---------------------------------------------------------------------------
__global__ __launch_bounds__(256) void zero_z_kernel(float* __restrict__ z) {
  z[blockIdx.x * 256 + threadIdx.x] = 0.0f;
}

// ---------------------------------------------------------------------------
// Kernel 1: z = spikes @ V   (split-K with fp32 atomics)
// grid = (BATCH/16, N_PRE/KCHUNK) = (32, 32), block = 256 (8 waves)
// ---------------------------------------------------------------------------
__global__ __launch_bounds__(256) void stage1_kernel(
    const float* __restrict__ spikes,   // [BATCH, N_PRE]
    const float* __restrict__ V,        // [N_PRE, RANK]
    float* __restrict__ z) {            // [BATCH, RANK] accumulator
  __shared__ float a_lds[16 * LDS_STRIDE];

  const int m0 = blockIdx.x * 16;       // batch-row tile
  const int k0 = blockIdx.y * KCHUNK;   // K chunk
  const int t  = threadIdx.x;           // 0..255

  // Cooperative, fully coalesced staging: 16 rows x 1024 cols of spikes.
  // Each thread moves float4; one row (4 KB) per iteration across the block.
#pragma unroll
  for (int i = 0; i < 16; ++i) {
    const float4 v = *(const float4*)(spikes + (size_t)(m0 + i) * N_PRE + k0 + t * 4);
    *(float4*)(&a_lds[i * LDS_STRIDE + t * 4]) = v;
  }
  __syncthreads();

  const int wave  = t >> 5;             // 0..7
  const int lane  = t & 31;
  const int ntile = wave & 3;           // 4 N-tiles cover RANK=64
  const int khalf = wave >> 2;          // split KCHUNK in 2 halves of 512
  const int n0    = ntile * 16;
  const int mlane = lane & 15;          // A: M = lane%16 ; B/C: N = lane%16
  const int kq    = (lane >> 4) * 2;    // A/B: K sub-offset = 2*(lane/16)

  // A fragment source (LDS): a[m, kq + kk + v], v=0,1 contiguous -> ds_load_b64
  const float* aRow = &a_lds[mlane * LDS_STRIDE + khalf * 512 + kq];
  // B fragment source (global, L2-resident): B(k,n) = V[k*RANK + n]
  //   lane supplies n = n0 + lane%16, rows k = k0 + khalf*512 + kk + kq + g
  const float* bBase = V + (size_t)(k0 + khalf * 512 + kq) * RANK + n0 + mlane;

  v8f c = {};
  for (int kk = 0; kk < 512; kk += 4) {
    v2f a = *(const v2f*)(aRow + kk);
    v2f b;
    b.x = bBase[(size_t)(kk + 0) * RANK];
    b.y = bBase[(size_t)(kk + 1) * RANK];
    c = wmma_f32_16x16x4(a, b, c);
  }

  // C/D layout: element g -> M = m0 + g + 8*(lane/16), N = n0 + lane%16.
  float* zcol = z + n0 + mlane;
  const int mBase = m0 + (lane >> 4) * 8;
#pragma unroll
  for (int g = 0; g < 8; ++g) {
    unsafeAtomicAdd(zcol + (size_t)(mBase + g) * RANK, c[g]);
  }
}

// ---------------------------------------------------------------------------
// Kernel 2: y = z @ U^T      B(k,n) = U[n*RANK + k]
// grid = (N_POST/128, BATCH/16) = (256, 32), block = 256 (8 waves, 16 cols each)
// ---------------------------------------------------------------------------
__global__ __launch_bounds__(256) void stage2_kernel(
    const float* __restrict__ z,        // [BATCH, RANK]
    const float* __restrict__ U,        // [N_POST, RANK]
    float* __restrict__ y) {            // [BATCH, N_POST]
  const int t    = threadIdx.x;
  const int wave = t >> 5;
  const int lane = t & 31;

  const int m0 = blockIdx.y * 16;
  const int n0 = blockIdx.x * 128 + wave * 16;
  const int mlane = lane & 15;
  const int kq    = (lane >> 4) * 2;

  // Preload all 16 A fragments (K=64): a[i] holds k = 4*i + kq + {0,1}
  v2f a[16];
  const float* zRow = z + (size_t)(m0 + mlane) * RANK + kq;
#pragma unroll
  for (int i = 0; i < 16; ++i) {
    a[i] = *(const v2f*)(zRow + i * 4);
  }

  // B fragments: lane n = n0 + lane%16 -> row of U; k = 4*i + kq + {0,1}
  const float* bRow = U + (size_t)(n0 + mlane) * RANK + kq;

  v8f c = {};
#pragma unroll
  for (int i = 0; i < 16; ++i) {
    v2f b = *(const v2f*)(bRow + i * 4);
    c = wmma_f32_16x16x4(a[i], b, c);
  }

  // Store: element g -> row m0 + g + 8*(lane/16), col n0 + lane%16.
  const int mBase = m0 + (lane >> 4) * 8;
  float* out = y + n0 + mlane;
#pragma unroll
  for (int g = 0; g < 8; ++g) {
    out[(size_t)(mBase + g) * N_POST] = c[g];
  }
}

// ---------------------------------------------------------------------------
extern "C" void kernel_launch(void* const* d_in, const int* in_sizes, int n_in,
                              void* d_out, int out_size, void* d_ws, size_t ws_size,
                              hipStream_t stream) {
  // setup_inputs() order: spikes, U, V, mask_row_ptr, mask_col_idx, mask_values
  const float* spikes = (const float*)d_in[0];
  const float* U      = (const float*)d_in[1];
  const float* V      = (const float*)d_in[2];
  float* y = (float*)d_out;
  float* z = (float*)d_ws;              // BATCH*RANK f32 = 128 KB scratch

  (void)in_sizes; (void)n_in; (void)out_size; (void)ws_size;

  // 1) zero the split-K accumulator
  zero_z_kernel<<<(BATCH * RANK) / 256, 256, 0, stream>>>(z);

  // 2) z = spikes @ V  (split-K over 32 chunks, fp32 atomics)
  dim3 g1(BATCH / 16, N_PRE / KCHUNK);
  stage1_kernel<<<g1, 256, 0, stream>>>(spikes, V, z);

  // 3) y = z @ U^T
  dim3 g2(N_POST / 128, BATCH / 16);
  stage2_kernel<<<g2, 256, 0, stream>>>(z, U, y);
}